// FastAlignDistribution_89489938580250
// MI455X (gfx1250) — compile-verified
//
#include <hip/hip_runtime.h>
#include <math.h>
#include <stdint.h>

#define BATCH   128
#define MMAX    512
#define NTOT    513            // N_MAX + 1
#define ESZ     8192           // E_SIZE (theta columns)
#define FROWS   8192           // F_SIZE (theta rows)
#define LOG2E   1.4426950408889634f

typedef float v4f __attribute__((ext_vector_type(4)));   // native clang vector
                                                         // (nontemporal-capable)

// ---------------------------------------------------------------------------
// Stage 1: rowLSE[r] = logsumexp(theta[r, :]) for r in [0, 8192).
// One 256-thread block per row; 32 floats per thread held in registers
// (8 x 128-bit coalesced loads), LDS tree reduce for max then sum.
// Streams theta (256 MB) exactly once (~11us at 23.3 TB/s). Loads are
// NON-TEMPORAL so the zero-reuse stream does not evict the 192MB L2 that
// stage 2 wants full of theta rows.
// ---------------------------------------------------------------------------
__global__ __launch_bounds__(256) void theta_row_lse_kernel(
    const float* __restrict__ theta, float* __restrict__ rlse) {
  const int row = blockIdx.x;
  const int t   = threadIdx.x;
  const float* rp = theta + (size_t)row * ESZ;

  v4f v[8];
#pragma unroll
  for (int c = 0; c < 8; ++c)
    v[c] = __builtin_nontemporal_load((const v4f*)(rp + c * 1024 + t * 4));

  float mx = -INFINITY;
#pragma unroll
  for (int c = 0; c < 8; ++c)
    mx = fmaxf(mx, fmaxf(fmaxf(v[c].x, v[c].y), fmaxf(v[c].z, v[c].w)));

  __shared__ float red[256];
  red[t] = mx;
  __syncthreads();
  for (int s = 128; s > 0; s >>= 1) {
    if (t < s) red[t] = fmaxf(red[t], red[t + s]);
    __syncthreads();
  }
  const float rowmax = red[0];
  __syncthreads();

  float sum = 0.f;
#pragma unroll
  for (int c = 0; c < 8; ++c) {
    sum += exp2f((v[c].x - rowmax) * LOG2E);
    sum += exp2f((v[c].y - rowmax) * LOG2E);
    sum += exp2f((v[c].z - rowmax) * LOG2E);
    sum += exp2f((v[c].w - rowmax) * LOG2E);
  }
  red[t] = sum;
  __syncthreads();
  for (int s = 128; s > 0; s >>= 1) {
    if (t < s) red[t] += red[t + s];
    __syncthreads();
  }
  if (t == 0) rlse[row] = rowmax + logf(red[0]);
}

// ---------------------------------------------------------------------------
// Stage 2: out[b,i] = log( p0*exp(tls0) + (1-p0)*accS/accLH )
//   accLH = sum_{j=1..512} exp(lam*H_ij)          (pure ALU)
//   accS  = sum_{j=1..512} exp(lam*H_ij + tls_ij) (one scattered gather per j)
// All summands <= 0 (log-probs), so direct exp2-sum is safe: no running max.
//
// ONE 512-thread block (16 waves) per batch b: all 512 e-columns of each
// theta row f[b,j] are gathered by a single block in one pass, so each of
// the 513 rows' cachelines is touched once per batch (vs twice with 2
// blocks/batch). f[b,:] is staged into LDS with the CDNA5 async
// global->LDS DMA path (ASYNCcnt), rlse[f[b,j]] is gathered once per block
// into LDS, and theta gathers are prefetched 24 iterations ahead with
// global_prefetch_b8.
// ---------------------------------------------------------------------------
__global__ __launch_bounds__(512) void align_lse_kernel(
    const int* __restrict__ e, const int* __restrict__ f,
    const int* __restrict__ m, const int* __restrict__ n,
    const float* __restrict__ theta, const float* __restrict__ lambd,
    const float* __restrict__ p0v, const float* __restrict__ rlse,
    float* __restrict__ out) {
  const int b = blockIdx.x;
  const int t = threadIdx.x;
  const int i = t;

  __shared__ int   s_f[NTOT];   // f[b, j]
  __shared__ float s_r[NTOT];   // rlse[f[b, j]]

  // --- Async-stage the contiguous f[b, 0..511] row into LDS: exactly one
  //     b32 op per lane, every wave fully active (no EXEC divergence on the
  //     async path). Tail element j=512 via the plain path.
  {
    unsigned lds0 = (unsigned)(uintptr_t)&s_f[t];
    int goff0 = (b * NTOT + t) * 4;
    asm volatile("global_load_async_to_lds_b32 %0, %1, %2"
                 :: "v"(lds0), "v"(goff0), "s"(f) : "memory");
    asm volatile("s_wait_asynccnt 0" ::: "memory");
  }
  if (t == 0) s_f[512] = f[b * NTOT + 512];
  __syncthreads();

  // --- Gather rlse[f[b,j]] once per block into LDS.
  for (int j = t; j < NTOT; j += 512) s_r[j] = rlse[s_f[j]];
  __syncthreads();

  // --- Per-thread scalars (b is block-uniform -> these scalarize).
  const int   ei    = e[b * MMAX + i];
  const float inv_m = 1.0f / (float)m[b];
  const float inv_n = 1.0f / (float)n[b];
  const float lam   = expf(lambd[0]);
  const float lam2  = lam * LOG2E;             // exp(lam*H) = exp2(-lam2*|.|)
  const float fi    = (float)i * inv_m;
  const float* tcol = theta + ei;              // fixed column, rows vary

  float accLH = 0.f, accS = 0.f;
#pragma unroll 4
  for (int j = 1; j < NTOT; ++j) {
    const int fj = s_f[j];
    const float x = tcol[(size_t)fj << 13];    // theta[f[b,j], e[b,i]]
    int jp = j + 24; jp = (jp > 512) ? 512 : jp;          // branchless clamp
    __builtin_prefetch(tcol + ((size_t)s_f[jp] << 13), 0, 1);
    const float h2 = -lam2 * fabsf(fi - (float)j * inv_n); // log2(e^{lam*H})
    accLH += exp2f(h2);
    accS  += exp2f(h2 + (x - s_r[j]) * LOG2E);
  }

  // --- j = 0 term and final combine.
  const float x0 = tcol[(size_t)s_f[0] << 13];
  const float e0 = exp2f((x0 - s_r[0]) * LOG2E);           // exp(tls_0)
  const float pv = p0v[0];
  const float p0 = 1.0f / (1.0f + expf(-pv));
  out[b * MMAX + i] = logf(p0 * e0 + (1.0f - p0) * (accS / accLH));
}

// ---------------------------------------------------------------------------
extern "C" void kernel_launch(void* const* d_in, const int* in_sizes, int n_in,
                              void* d_out, int out_size, void* d_ws, size_t ws_size,
                              hipStream_t stream) {
  (void)in_sizes; (void)n_in; (void)out_size; (void)ws_size;
  const int*   e     = (const int*)d_in[0];
  const int*   f     = (const int*)d_in[1];
  const int*   m     = (const int*)d_in[2];
  const int*   n     = (const int*)d_in[3];
  const float* theta = (const float*)d_in[4];
  const float* lambd = (const float*)d_in[5];
  const float* p0    = (const float*)d_in[6];
  float* out  = (float*)d_out;
  float* rlse = (float*)d_ws;                  // 8192 floats = 32 KB scratch

  theta_row_lse_kernel<<<FROWS, 256, 0, stream>>>(theta, rlse);

  align_lse_kernel<<<BATCH, 512, 0, stream>>>(e, f, m, n, theta, lambd, p0,
                                              rlse, out);
}